// CellWS_22703197126758
// MI455X (gfx1250) — compile-verified
//
#include <hip/hip_runtime.h>
#include <hip/hip_bf16.h>

#define GN 50000
#define GD 64
#define GE 800000
#define NEG_SLOPE 0.2f

typedef float v2f __attribute__((ext_vector_type(2)));
typedef float v8f __attribute__((ext_vector_type(8)));

__device__ __forceinline__ float lrelu(float x) { return x > 0.f ? x : NEG_SLOPE * x; }
__device__ __forceinline__ unsigned fenc(float f) {
    unsigned u = __float_as_uint(f);
    return (u >> 31) ? ~u : (u | 0x80000000u);
}
__device__ __forceinline__ float fdec(unsigned u) {
    return (u & 0x80000000u) ? __uint_as_float(u & 0x7fffffffu) : __uint_as_float(~u);
}

// ---------------- per-layer init: zero accumulators, relu h if needed -------
__global__ void k_layer_init(const float* __restrict__ hprev, float* __restrict__ hcur,
                             float* __restrict__ gat, float* __restrict__ gcn,
                             float* __restrict__ agg, float* __restrict__ deg,
                             float* __restrict__ wsum, float* __restrict__ denom,
                             int do_relu) {
    int t = blockIdx.x * blockDim.x + threadIdx.x;
    if (t < GN * GD) {
        gat[t] = 0.f; gcn[t] = 0.f; agg[t] = 0.f;
        if (do_relu) hcur[t] = fmaxf(hprev[t], 0.f);
    }
    if (t < GN) { deg[t] = 1.f; wsum[t] = 0.f; denom[t] = 0.f; }
}

// ---------------- edge weights + degree sums --------------------------------
__global__ void k_edge_w(const float* __restrict__ logit, const int* __restrict__ dst,
                         float* __restrict__ ew, float* __restrict__ deg,
                         float* __restrict__ wsum) {
    int e = blockIdx.x * blockDim.x + threadIdx.x;
    if (e >= GE) return;
    float w = 1.f / (1.f + expf(-logit[e]));
    ew[e] = w;
    int d = dst[e];
    atomicAdd(&deg[d], w);
    atomicAdd(&wsum[d], w);
}

__global__ void k_node_dinv(const float* __restrict__ deg, float* __restrict__ dinv) {
    int n = blockIdx.x * blockDim.x + threadIdx.x;
    if (n < GN) dinv[n] = rsqrtf(deg[n]);  // deg >= 1 always (self loop)
}

// ---------------- dual-B WMMA GEMM: C1 = A@B1, C2 = A@B2  ([N,64]x[64,64]) --
__global__ void k_gemm2(const float* __restrict__ A,
                        const float* __restrict__ B1, float* __restrict__ C1,
                        const float* __restrict__ B2, float* __restrict__ C2) {
    int lane = threadIdx.x & 31;
    int wv   = threadIdx.x >> 5;       // 4 waves -> 4 column tiles
    int c0   = wv * 16;
    int r0   = blockIdx.x * 16;
    int half = lane >> 4;
    int lr   = lane & 15;
    int arow = r0 + lr;

    v8f acc1 = {0.f,0.f,0.f,0.f,0.f,0.f,0.f,0.f};
    v8f acc2 = {0.f,0.f,0.f,0.f,0.f,0.f,0.f,0.f};
#pragma unroll
    for (int k0 = 0; k0 < 64; k0 += 4) {
        int ka = k0 + half * 2;
        v2f a; a.x = A[arow * 64 + ka]; a.y = A[arow * 64 + ka + 1];
        v2f b; b.x = B1[ka * 64 + c0 + lr]; b.y = B1[(ka + 1) * 64 + c0 + lr];
        acc1 = __builtin_amdgcn_wmma_f32_16x16x4_f32(false, a, false, b, (short)0, acc1, false, false);
        v2f c; c.x = B2[ka * 64 + c0 + lr]; c.y = B2[(ka + 1) * 64 + c0 + lr];
        acc2 = __builtin_amdgcn_wmma_f32_16x16x4_f32(false, a, false, c, (short)0, acc2, false, false);
    }
#pragma unroll
    for (int r = 0; r < 8; r++) {
        int m = r0 + r + half * 8;
        C1[m * 64 + c0 + lr] = acc1[r];
        C2[m * 64 + c0 + lr] = acc2[r];
    }
}

// ---------------- GAT per-node scores + emax init with self-loop ------------
__global__ void k_gat_scores(const float* __restrict__ hA, const float* __restrict__ asrc,
                             const float* __restrict__ adst, float* __restrict__ es,
                             float* __restrict__ ed, unsigned* __restrict__ emax) {
    int n = blockIdx.x * blockDim.x + threadIdx.x;
    if (n >= GN) return;
    float s = 0.f, d = 0.f;
#pragma unroll 8
    for (int c = 0; c < 64; c++) {
        float v = hA[n * 64 + c];
        s += v * asrc[c];
        d += v * adst[c];
    }
    es[n] = s; ed[n] = d;
    emax[n] = fenc(lrelu(s + d));      // self-loop seeds the segment max
}

__global__ void k_edge_max(const int* __restrict__ src, const int* __restrict__ dst,
                           const float* __restrict__ es, const float* __restrict__ ed,
                           float* __restrict__ ebuf, unsigned* __restrict__ emax) {
    int e = blockIdx.x * blockDim.x + threadIdx.x;
    if (e >= GE) return;
    float ev = lrelu(es[src[e]] + ed[dst[e]]);
    ebuf[e] = ev;
    atomicMax(&emax[dst[e]], fenc(ev));
}

__global__ void k_edge_exp(const int* __restrict__ dst, const unsigned* __restrict__ emax,
                           float* __restrict__ ebuf, float* __restrict__ denom) {
    int e = blockIdx.x * blockDim.x + threadIdx.x;
    if (e >= GE) return;
    int d = dst[e];
    float p = expf(ebuf[e] - fdec(emax[d]));
    ebuf[e] = p;
    atomicAdd(&denom[d], p);
}

__global__ void k_node_soft(const float* __restrict__ es, const float* __restrict__ ed,
                            const unsigned* __restrict__ emax, float* __restrict__ denom,
                            float* __restrict__ selfA, float* __restrict__ wsum) {
    int n = blockIdx.x * blockDim.x + threadIdx.x;
    if (n >= GN) return;
    float p = expf(lrelu(es[n] + ed[n]) - fdec(emax[n]));
    float den = denom[n] + p;          // include self-loop term
    denom[n] = den;
    selfA[n] = p / den;
    wsum[n]  = 1.f / fmaxf(wsum[n], 1e-6f);   // becomes SAGE 1/mean-weight
}

// ---------------- fused vector scatter: GAT + GCN + agg ---------------------
__global__ void k_edge_scatter(const int* __restrict__ src, const int* __restrict__ dst,
                               const float* __restrict__ ew, const float* __restrict__ ebuf,
                               const float* __restrict__ denom, const float* __restrict__ dinv,
                               const float* __restrict__ h, const float* __restrict__ hg,
                               const float* __restrict__ hA, float* __restrict__ gat,
                               float* __restrict__ gcn, float* __restrict__ agg) {
    long long t = (long long)blockIdx.x * blockDim.x + threadIdx.x;
    if (t >= (long long)GE * 64) return;
    int e = (int)(t >> 6);
    int c = (int)(t & 63);
    int s = src[e], d = dst[e];
    float w     = ew[e];
    float alpha = ebuf[e] / denom[d];
    float nm    = dinv[s] * w * dinv[d];
    atomicAdd(&gat[d * 64 + c], alpha * hA[s * 64 + c]);
    atomicAdd(&gcn[d * 64 + c], nm * hg[s * 64 + c]);
    atomicAdd(&agg[d * 64 + c], w * h[s * 64 + c]);
}

// ---------------- combine weight matrices with mixture coefficients ---------
__global__ void k_wcomb(const float* __restrict__ w5, const float* __restrict__ eps,
                        const float* __restrict__ ginW, const float* __restrict__ ginB,
                        const float* __restrict__ sWl, const float* __restrict__ sWr,
                        const float* __restrict__ sB, const float* __restrict__ linW,
                        const float* __restrict__ linB, const float* __restrict__ gatB,
                        const float* __restrict__ gcnB, float* __restrict__ Wc1,
                        float* __restrict__ Wc2, float* __restrict__ Wc3,
                        float* __restrict__ bc) {
    int t = blockIdx.x * blockDim.x + threadIdx.x;
    if (t >= 64 * 64) return;
    float w0 = w5[0], w1 = w5[1], w2 = w5[2], w3 = w5[3], w4 = w5[4];
    float e1 = 1.f + eps[0];
    Wc1[t] = w2 * e1 * ginW[t] + w3 * sWl[t] + w4 * linW[t];
    Wc2[t] = w2 * ginW[t];
    Wc3[t] = w3 * sWr[t];
    if (t < 64)
        bc[t] = w0 * gatB[t] + w1 * gcnB[t] + w2 * ginB[t] + w3 * sB[t] + w4 * linB[t];
}

// ---------------- final: 3 WMMA chains + scatter parts + biases -------------
__global__ void k_final(const float* __restrict__ h, const float* __restrict__ agg,
                        const float* __restrict__ rw, const float* __restrict__ Wc1,
                        const float* __restrict__ Wc2, const float* __restrict__ Wc3,
                        const float* __restrict__ bc, const float* __restrict__ gat,
                        const float* __restrict__ gcn, const float* __restrict__ hA,
                        const float* __restrict__ hg, const float* __restrict__ selfA,
                        const float* __restrict__ dinv, const float* __restrict__ w5,
                        float* __restrict__ out) {
    int lane = threadIdx.x & 31;
    int wv   = threadIdx.x >> 5;
    int c0   = wv * 16;
    int r0   = blockIdx.x * 16;
    int half = lane >> 4;
    int lr   = lane & 15;
    int arow = r0 + lr;
    float rwv = rw[arow];

    v8f acc = {0.f,0.f,0.f,0.f,0.f,0.f,0.f,0.f};
#pragma unroll
    for (int k0 = 0; k0 < 64; k0 += 4) {   // h @ Wc1
        int ka = k0 + half * 2;
        v2f a; a.x = h[arow * 64 + ka]; a.y = h[arow * 64 + ka + 1];
        v2f b; b.x = Wc1[ka * 64 + c0 + lr]; b.y = Wc1[(ka + 1) * 64 + c0 + lr];
        acc = __builtin_amdgcn_wmma_f32_16x16x4_f32(false, a, false, b, (short)0, acc, false, false);
    }
#pragma unroll
    for (int k0 = 0; k0 < 64; k0 += 4) {   // agg @ Wc2
        int ka = k0 + half * 2;
        v2f a; a.x = agg[arow * 64 + ka]; a.y = agg[arow * 64 + ka + 1];
        v2f b; b.x = Wc2[ka * 64 + c0 + lr]; b.y = Wc2[(ka + 1) * 64 + c0 + lr];
        acc = __builtin_amdgcn_wmma_f32_16x16x4_f32(false, a, false, b, (short)0, acc, false, false);
    }
#pragma unroll
    for (int k0 = 0; k0 < 64; k0 += 4) {   // (agg * rw) @ Wc3   (SAGE mean)
        int ka = k0 + half * 2;
        v2f a; a.x = agg[arow * 64 + ka] * rwv; a.y = agg[arow * 64 + ka + 1] * rwv;
        v2f b; b.x = Wc3[ka * 64 + c0 + lr]; b.y = Wc3[(ka + 1) * 64 + c0 + lr];
        acc = __builtin_amdgcn_wmma_f32_16x16x4_f32(false, a, false, b, (short)0, acc, false, false);
    }

    float w0 = w5[0], w1 = w5[1];
#pragma unroll
    for (int r = 0; r < 8; r++) {
        int m = r0 + r + half * 8;
        int col = c0 + lr;
        float dv = dinv[m];
        float v = acc[r]
                + w0 * (gat[m * 64 + col] + selfA[m] * hA[m * 64 + col])
                + w1 * (gcn[m * 64 + col] + dv * dv * hg[m * 64 + col])
                + bc[col];
        out[m * 64 + col] = v;
    }
}

// ---------------------------------------------------------------------------
extern "C" void kernel_launch(void* const* d_in, const int* in_sizes, int n_in,
                              void* d_out, int out_size, void* d_ws, size_t ws_size,
                              hipStream_t stream) {
    const float* x       = (const float*)d_in[0];
    const int*   ei      = (const int*)d_in[1];
    const float* logits  = (const float*)d_in[2];
    const float* weights = (const float*)d_in[3];
    const float* gcnW    = (const float*)d_in[4];
    const float* gcnB    = (const float*)d_in[5];
    const float* gatW    = (const float*)d_in[6];
    const float* asrc    = (const float*)d_in[7];
    const float* adst    = (const float*)d_in[8];
    const float* gatB    = (const float*)d_in[9];
    const float* ginW    = (const float*)d_in[10];
    const float* ginB    = (const float*)d_in[11];
    const float* ginE    = (const float*)d_in[12];
    const float* sWl     = (const float*)d_in[13];
    const float* sWr     = (const float*)d_in[14];
    const float* sB      = (const float*)d_in[15];
    const float* linW    = (const float*)d_in[16];
    const float* linB    = (const float*)d_in[17];

    char* base = (char*)d_ws;
    size_t off = 0;
    auto carve = [&](size_t bytes) -> char* {
        char* p = base + off;
        off += (bytes + 255) & ~(size_t)255;
        return p;
    };
    const size_t ND = (size_t)GN * GD * sizeof(float);
    float*    hbuf   = (float*)carve(ND);
    float*    hnext  = (float*)carve(ND);
    float*    hg     = (float*)carve(ND);
    float*    hA     = (float*)carve(ND);
    float*    gat    = (float*)carve(ND);
    float*    gcn    = (float*)carve(ND);
    float*    agg    = (float*)carve(ND);
    float*    deg    = (float*)carve(GN * sizeof(float));
    float*    dinv   = (float*)carve(GN * sizeof(float));
    float*    wsum   = (float*)carve(GN * sizeof(float));
    float*    es     = (float*)carve(GN * sizeof(float));
    float*    ed     = (float*)carve(GN * sizeof(float));
    unsigned* emax   = (unsigned*)carve(GN * sizeof(unsigned));
    float*    denom  = (float*)carve(GN * sizeof(float));
    float*    selfA  = (float*)carve(GN * sizeof(float));
    float*    ew     = (float*)carve((size_t)GE * sizeof(float));
    float*    ebuf   = (float*)carve((size_t)GE * sizeof(float));
    float*    Wc1    = (float*)carve(64 * 64 * sizeof(float));
    float*    Wc2    = (float*)carve(64 * 64 * sizeof(float));
    float*    Wc3    = (float*)carve(64 * 64 * sizeof(float));
    float*    bc     = (float*)carve(64 * sizeof(float));
    (void)ws_size; (void)in_sizes; (void)n_in; (void)out_size;

    const int gridND = (GN * GD + 255) / 256;
    const int gridN  = (GN + 255) / 256;
    const int gridE  = (GE + 255) / 256;
    const int gridEV = (int)(((long long)GE * 64 + 255) / 256);
    const int gridT  = GN / 16;  // 50000 divisible by 16

    for (int i = 0; i < 3; i++) {
        const float* h   = (i == 0) ? x : hbuf;
        const int*   src = ei + (size_t)i * 2 * GE;
        const int*   dst = src + GE;

        k_layer_init<<<gridND, 256, 0, stream>>>(hnext, hbuf, gat, gcn, agg,
                                                 deg, wsum, denom, i > 0 ? 1 : 0);
        k_edge_w<<<gridE, 256, 0, stream>>>(logits + (size_t)i * GE, dst, ew, deg, wsum);
        k_node_dinv<<<gridN, 256, 0, stream>>>(deg, dinv);
        k_gemm2<<<gridT, 128, 0, stream>>>(h, gcnW + (size_t)i * 4096, hg,
                                           gatW + (size_t)i * 4096, hA);
        k_gat_scores<<<gridN, 256, 0, stream>>>(hA, asrc + (size_t)i * 64,
                                                adst + (size_t)i * 64, es, ed, emax);
        k_edge_max<<<gridE, 256, 0, stream>>>(src, dst, es, ed, ebuf, emax);
        k_edge_exp<<<gridE, 256, 0, stream>>>(dst, emax, ebuf, denom);
        k_node_soft<<<gridN, 256, 0, stream>>>(es, ed, emax, denom, selfA, wsum);
        k_edge_scatter<<<gridEV, 256, 0, stream>>>(src, dst, ew, ebuf, denom, dinv,
                                                   h, hg, hA, gat, gcn, agg);
        k_wcomb<<<16, 256, 0, stream>>>(weights + (size_t)i * 5, ginE + i,
                                        ginW + (size_t)i * 4096, ginB + (size_t)i * 64,
                                        sWl + (size_t)i * 4096, sWr + (size_t)i * 4096,
                                        sB + (size_t)i * 64, linW + (size_t)i * 4096,
                                        linB + (size_t)i * 64, gatB + (size_t)i * 64,
                                        gcnB + (size_t)i * 64, Wc1, Wc2, Wc3, bc);
        float* outp = (i == 2) ? (float*)d_out : hnext;
        k_final<<<gridT, 128, 0, stream>>>(h, agg, wsum, Wc1, Wc2, Wc3, bc,
                                           gat, gcn, hA, hg, selfA, dinv,
                                           weights + (size_t)i * 5, outp);
    }
}